// SwitchingDevGRUCell_75239237091518
// MI455X (gfx1250) — compile-verified
//
#include <hip/hip_runtime.h>

// SwitchingDevGRUCell on gfx1250 (MI455X), wave32 + V_WMMA_F32_16X16X4_F32.
//
// Shapes: B=32768, C=8, H=128, R=16, 3H=384.
// rec = x-masked dense low-rank:  Vh_all = h @ Vflat  (Vflat[k][c*16+r] = V[c,k,r])
//                                 Masked[b, c*16+r] = x[b,c] * Vh_all[b, c*16+r]
//                                 rec    = Masked' @ Uflat^T, with K extended by 8
//                                 rows of x[b,c] to fold the bias GEMMs in.
// Then the GRU elementwise pass entirely in registers.

typedef float v2f __attribute__((ext_vector_type(2)));
typedef float v8f __attribute__((ext_vector_type(8)));

#define B_TOT   32768
#define C_N     8
#define H_N     128
#define R_N     16
#define G3      384          // 3*H
#define KM      136          // 128 (V·h) + 8 (x columns for bias folding)
#define LDS_STR 137          // row stride in floats (odd -> conflict-free A reads)
#define WAVES   4
#define ROWS_W  16           // batch rows per wave (one 16xN WMMA tile)

__device__ __forceinline__ v8f wmma_f32(v2f a, v2f b, v8f c) {
    // 8 args: (neg_a, A, neg_b, B, c_mod, C, reuse_a, reuse_b)
    return __builtin_amdgcn_wmma_f32_16x16x4_f32(false, a, false, b, (short)0, c,
                                                 false, false);
}

__global__ __launch_bounds__(WAVES * 32) void
switching_gru_wmma_kernel(const float* __restrict__ x,    // [B, 8]
                          const float* __restrict__ h,    // [B, 128]
                          const float* __restrict__ U,    // [8, 384, 16]
                          const float* __restrict__ V,    // [8, 128, 16]
                          const float* __restrict__ bih,  // [8, 384]
                          const float* __restrict__ bhh,  // [8, 384]
                          float* __restrict__ out)        // [B, 128]
{
    __shared__ float lds[WAVES * ROWS_W * LDS_STR];

    const int lane = threadIdx.x & 31;
    const int wave = threadIdx.x >> 5;
    const int hf   = lane >> 4;      // lane half: selects K pair / M half
    const int l16  = lane & 15;

    const int row0 = blockIdx.x * (WAVES * ROWS_W) + wave * ROWS_W;
    float* mlds = &lds[wave * ROWS_W * LDS_STR];

    // ---------------- A fragments of h tile (K = 128) ----------------
    // A layout (16x4 f32): lane = M (two halves), VGPR0/1 = K, K+1; halves hold K, K+2.
    v2f a[KM / 4];  // 34 fragments; [32],[33] filled from LDS later (x columns)
    {
        const float* hrow = h + (size_t)(row0 + l16) * H_N;
        #pragma unroll
        for (int kk = 0; kk < 32; ++kk) {
            const int k = kk * 4 + hf * 2;
            a[kk] = *(const v2f*)&hrow[k];
        }
    }

    // ---------------- GEMM1 + mask, staged to LDS ----------------
    // B layout (4x16 f32): lane = N (two halves), halves hold K rows {0,1}/{2,3}.
    #pragma unroll 1
    for (int c = 0; c < C_N; ++c) {
        const float* Vc = V + c * (H_N * R_N);
        v8f acc = {};
        #pragma unroll
        for (int kk = 0; kk < 32; ++kk) {
            const int k = kk * 4 + hf * 2;
            v2f b;
            b.x = Vc[k * R_N + l16];
            b.y = Vc[(k + 1) * R_N + l16];
            acc = wmma_f32(a[kk], b, acc);
        }
        // C/D layout: VGPR j, lanes 0-15 -> M=j, lanes 16-31 -> M=j+8; N = l16.
        #pragma unroll
        for (int j = 0; j < 8; ++j) {
            const int mrow = j + hf * 8;
            const float m  = x[(size_t)(row0 + mrow) * C_N + c];
            mlds[mrow * LDS_STR + c * 16 + l16] = acc[j] * m;
        }
    }
    // x columns -> Masked'[:, 128..135]
    for (int idx = lane; idx < ROWS_W * C_N; idx += 32) {
        const int mrow = idx >> 3, c = idx & 7;
        mlds[mrow * LDS_STR + 128 + c] = x[(size_t)(row0 + mrow) * C_N + c];
    }
    __syncthreads();

    // ---------------- reload A fragments of Masked' (K = 136) ----------------
    #pragma unroll
    for (int kk = 0; kk < KM / 4; ++kk) {
        const int k = kk * 4 + hf * 2;
        v2f t;
        t.x = mlds[l16 * LDS_STR + k];
        t.y = mlds[l16 * LDS_STR + k + 1];
        a[kk] = t;
    }

    // ---------------- GEMM2 per 16-column output tile + GRU ----------------
    #pragma unroll 1
    for (int t = 0; t < 8; ++t) {
        v8f ar = {}, az = {}, an = {}, ani = {};
        const int g = t * 16 + l16;  // output column within H

        #pragma unroll
        for (int kk = 0; kk < 32; ++kk) {
            const int k = kk * 4 + hf * 2;
            const int c = k >> 4;
            const int r = k & 15;    // r,r+1 contiguous in U -> float2 load
            const float* Uc = U + c * (G3 * R_N);
            const v2f br = *(const v2f*)&Uc[(g)       * R_N + r];
            const v2f bz = *(const v2f*)&Uc[(128 + g) * R_N + r];
            const v2f bn = *(const v2f*)&Uc[(256 + g) * R_N + r];
            ar = wmma_f32(a[kk], br, ar);
            az = wmma_f32(a[kk], bz, az);
            an = wmma_f32(a[kk], bn, an);
        }
        // bias rows (K = 128..135): A holds x[b,c]; r/z fold bih+bhh, n keeps them split.
        #pragma unroll
        for (int kk = 32; kk < 34; ++kk) {
            const int c0 = (kk - 32) * 4 + hf * 2;
            v2f br, bz, bn, bni;
            br.x  = bhh[c0 * G3 + g]             + bih[c0 * G3 + g];
            br.y  = bhh[(c0 + 1) * G3 + g]       + bih[(c0 + 1) * G3 + g];
            bz.x  = bhh[c0 * G3 + 128 + g]       + bih[c0 * G3 + 128 + g];
            bz.y  = bhh[(c0 + 1) * G3 + 128 + g] + bih[(c0 + 1) * G3 + 128 + g];
            bn.x  = bhh[c0 * G3 + 256 + g];
            bn.y  = bhh[(c0 + 1) * G3 + 256 + g];
            bni.x = bih[c0 * G3 + 256 + g];
            bni.y = bih[(c0 + 1) * G3 + 256 + g];
            ar  = wmma_f32(a[kk], br,  ar);
            az  = wmma_f32(a[kk], bz,  az);
            an  = wmma_f32(a[kk], bn,  an);
            ani = wmma_f32(a[kk], bni, ani);
        }

        // GRU elementwise on C/D-layout registers, store out tile.
        #pragma unroll
        for (int j = 0; j < 8; ++j) {
            const int brow = row0 + j + hf * 8;
            const float hv = h[(size_t)brow * H_N + g];
            const float rr = 1.0f / (1.0f + __expf(-ar[j]));
            const float zz = 1.0f / (1.0f + __expf(-az[j]));
            const float nn = tanhf(ani[j] + rr * an[j]);
            out[(size_t)brow * H_N + g] = (1.0f - zz) * nn + zz * hv;
        }
    }
}

extern "C" void kernel_launch(void* const* d_in, const int* in_sizes, int n_in,
                              void* d_out, int out_size, void* d_ws, size_t ws_size,
                              hipStream_t stream) {
    (void)in_sizes; (void)n_in; (void)out_size; (void)d_ws; (void)ws_size;
    const float* x   = (const float*)d_in[0];
    const float* h   = (const float*)d_in[1];
    const float* U   = (const float*)d_in[2];
    const float* V   = (const float*)d_in[3];
    const float* bih = (const float*)d_in[4];
    const float* bhh = (const float*)d_in[5];
    float* out = (float*)d_out;

    dim3 block(WAVES * 32);                   // 128 threads = 4 waves
    dim3 grid(B_TOT / (WAVES * ROWS_W));      // 512 workgroups x 64 rows
    switching_gru_wmma_kernel<<<grid, block, 0, stream>>>(x, h, U, V, bih, bhh, out);
}